// Memory_75763223101792
// MI455X (gfx1250) — compile-verified
//
#include <hip/hip_runtime.h>
#include <math.h>

// ---------------- problem constants (from reference) ----------------
#define B_STEPS 1024
#define DK 1024
#define DU 1024
#define MCOLS 8192
// persistent scan grid
#define NBLK 128
#define TPB 256
#define COLS_PER_BLK (MCOLS / NBLK)            // 64
#define NWAVES (TPB / 32)                      // 8 waves (wave32)
#define COLS_PER_WAVE (COLS_PER_BLK / NWAVES)  // 8

typedef float v2f __attribute__((ext_vector_type(2)));
typedef float v8f __attribute__((ext_vector_type(8)));

// ---------------- device-scope atomic helpers ----------------
__device__ __forceinline__ float gloadf(const float* p) {
  return __hip_atomic_load(p, __ATOMIC_RELAXED, __HIP_MEMORY_SCOPE_AGENT);
}
__device__ __forceinline__ void gstoref(float* p, float v) {
  __hip_atomic_store(p, v, __ATOMIC_RELAXED, __HIP_MEMORY_SCOPE_AGENT);
}

// Grid-wide split barrier for the persistent scan kernel (128 resident WGs).
__device__ __forceinline__ void grid_sync(int* cnt, int* gen) {
  __syncthreads();
  if (threadIdx.x == 0) {
    int g = __hip_atomic_load(gen, __ATOMIC_ACQUIRE, __HIP_MEMORY_SCOPE_AGENT);
    int a = __hip_atomic_fetch_add(cnt, 1, __ATOMIC_ACQ_REL, __HIP_MEMORY_SCOPE_AGENT);
    if (a == NBLK - 1) {
      __hip_atomic_store(cnt, 0, __ATOMIC_RELAXED, __HIP_MEMORY_SCOPE_AGENT);
      __hip_atomic_fetch_add(gen, 1, __ATOMIC_ACQ_REL, __HIP_MEMORY_SCOPE_AGENT);
    } else {
      while (__hip_atomic_load(gen, __ATOMIC_ACQUIRE, __HIP_MEMORY_SCOPE_AGENT) == g)
        __builtin_amdgcn_s_sleep(1);
    }
  }
  __syncthreads();
}

// ---------------- prep kernels ----------------
// [D, M] row-major -> [M, D] so each memory column is a contiguous 4KB stream.
__global__ void __launch_bounds__(TPB) transpose_kernel(
    const float* __restrict__ src, float* __restrict__ dst, int D, int M) {
  size_t i = (size_t)blockIdx.x * TPB + threadIdx.x;
  if (i < (size_t)D * M) {
    int d = (int)(i / M);
    int m = (int)(i % M);
    dst[(size_t)m * D + d] = src[i];
  }
}

// ||k_t|| for every step, precomputed once.
__global__ void __launch_bounds__(TPB) knorm_kernel(
    const float* __restrict__ k, float* __restrict__ knorm) {
  __shared__ float red[TPB];
  const int t = blockIdx.x;
  float s = 0.f;
  for (int i = threadIdx.x; i < DK; i += TPB) {
    float v = k[(size_t)t * DK + i];
    s += v * v;
  }
  red[threadIdx.x] = s;
  __syncthreads();
  for (int o = TPB / 2; o > 0; o >>= 1) {
    if (threadIdx.x < o) red[threadIdx.x] += red[threadIdx.x + o];
    __syncthreads();
  }
  if (threadIdx.x == 0) knorm[t] = sqrtf(red[0]);
}

// ---------------- persistent fused-scan kernel ----------------
// Iteration g: apply rank-1 decay update for step g-1 (g>0) and, with the
// post-update values still in registers, compute step-g similarities, exp
// sums and both u_mem partial variants (unscaled / column-normalized).
// Column renormalization is LAZY: cosine sim is scale-invariant, so only
// per-column scale factors (LDS) are updated; they fold into the next decay.
// One memory pass + one grid barrier per step. Accumulators are mod-3
// parity buffered so recycling never races the single barrier.
__global__ void __launch_bounds__(TPB) scan_kernel(
    const float* __restrict__ k, const float* __restrict__ u,
    float* __restrict__ memKT, float* __restrict__ memUT,
    float* __restrict__ u_mem, const float* __restrict__ knorm,
    int* cnt, int* gen, float* Sr, float* Sw, float* frob,
    float* upA_G, float* upB_G) {
  __shared__ __align__(16) float kc_lds[DK];   // k[g-1] (update)
  __shared__ __align__(16) float uc_lds[DU];   // u[g-1] (update)
  __shared__ __align__(16) float kn_lds[DK];   // k[g]   (next sim)
  __shared__ __align__(16) float upA_lds[DU];  // u_mem partial, unscaled
  __shared__ __align__(16) float upB_lds[DU];  // u_mem partial, col-normalized
  __shared__ float ew_lds[COLS_PER_BLK];
  __shared__ float cnk_lds[COLS_PER_BLK];
  __shared__ float cnu_lds[COLS_PER_BLK];
  __shared__ float sK_lds[COLS_PER_BLK];       // lazy column scales
  __shared__ float sU_lds[COLS_PER_BLK];
  __shared__ float sS[2];
  __shared__ float sF;
  __shared__ float Sw_sh;

  const int tid = threadIdx.x;
  const int lane = tid & 31;
  const int wave = tid >> 5;
  const int col0 = blockIdx.x * COLS_PER_BLK;

  if (tid < COLS_PER_BLK) { sK_lds[tid] = 1.f; sU_lds[tid] = 1.f; }

  for (int g = 0; g < B_STEPS; ++g) {
    const int q = g % 3;        // this iteration's accumulator parity
    const int z = (g + 1) % 3;  // parity safe to recycle now

    for (int i = tid; i < DK; i += TPB) kn_lds[i] = k[(size_t)g * DK + i];
    if (g > 0) {
      for (int i = tid; i < DK; i += TPB) kc_lds[i] = k[(size_t)(g - 1) * DK + i];
      for (int i = tid; i < DU; i += TPB) uc_lds[i] = u[(size_t)(g - 1) * DU + i];
    }
    for (int i = tid; i < DU; i += TPB) { upA_lds[i] = 0.f; upB_lds[i] = 0.f; }
    if (tid == 0) { sS[0] = 0.f; sS[1] = 0.f; sF = 0.f; }
    if (blockIdx.x == 0) {  // recycle parity z (last read 2 barriers ago)
      if (tid == 0) { gstoref(&Sr[z], 0.f); gstoref(&Sw[z], 0.f); gstoref(&frob[z], 0.f); }
      for (int i = tid; i < DU; i += TPB) {
        gstoref(&upA_G[z * DU + i], 0.f);
        gstoref(&upB_G[z * DU + i], 0.f);
      }
    }
    __syncthreads();

    const float knrm = knorm[g];
    const float invSw = (g > 0) ? (0.1f / Sw_sh) : 0.f;  // LR folded in
    const float4* kc4 = (const float4*)kc_lds;
    const float4* uc4 = (const float4*)uc_lds;
    const float4* kn4 = (const float4*)kn_lds;

    float4 upA[8], upB[8];
#pragma unroll
    for (int i = 0; i < 8; ++i) {
      upA[i] = make_float4(0.f, 0.f, 0.f, 0.f);
      upB[i] = make_float4(0.f, 0.f, 0.f, 0.f);
    }

    for (int c = 0; c < COLS_PER_WAVE; ++c) {
      const int li = wave * COLS_PER_WAVE + c;
      const int m = col0 + li;
      float4* pk = (float4*)(memKT + (size_t)m * DK);
      float4* pu = (float4*)(memUT + (size_t)m * DU);
      if (c + 1 < COLS_PER_WAVE) {
        __builtin_prefetch(memKT + (size_t)(m + 1) * DK, 0, 1);
        __builtin_prefetch(memUT + (size_t)(m + 1) * DU, 0, 1);
      }
      float4 nu[8];
      float cnk = 0.f, cnu = 0.f, dot = 0.f;
      if (g > 0) {
        const float a  = 0.99f * sK_lds[li];   // decay x lazy scale
        const float au = 0.99f * sU_lds[li];
        const float w  = ew_lds[li] * invSw;   // write weight (LR/Sw folded)
#pragma unroll
        for (int i2 = 0; i2 < 8; ++i2) {
          const int idx = i2 * 32 + lane;
          float4 v = pk[idx];
          float4 kcv = kc4[idx];
          float4 knv = kn4[idx];
          v.x = a * v.x + w * kcv.x; v.y = a * v.y + w * kcv.y;
          v.z = a * v.z + w * kcv.z; v.w = a * v.w + w * kcv.w;
          pk[idx] = v;
          cnk += v.x * v.x + v.y * v.y + v.z * v.z + v.w * v.w;
          dot += v.x * knv.x + v.y * knv.y + v.z * knv.z + v.w * knv.w;
          float4 vu = pu[idx];
          float4 ucv = uc4[idx];
          vu.x = au * vu.x + w * ucv.x; vu.y = au * vu.y + w * ucv.y;
          vu.z = au * vu.z + w * ucv.z; vu.w = au * vu.w + w * ucv.w;
          pu[idx] = vu;
          cnu += vu.x * vu.x + vu.y * vu.y + vu.z * vu.z + vu.w * vu.w;
          nu[i2] = vu;
        }
      } else {  // prologue: read-only pass over the initial memory
#pragma unroll
        for (int i2 = 0; i2 < 8; ++i2) {
          const int idx = i2 * 32 + lane;
          float4 v = pk[idx];
          float4 knv = kn4[idx];
          cnk += v.x * v.x + v.y * v.y + v.z * v.z + v.w * v.w;
          dot += v.x * knv.x + v.y * knv.y + v.z * knv.z + v.w * knv.w;
          float4 vu = pu[idx];
          cnu += vu.x * vu.x + vu.y * vu.y + vu.z * vu.z + vu.w * vu.w;
          nu[i2] = vu;
        }
      }
#pragma unroll
      for (int o = 16; o > 0; o >>= 1) {
        dot += __shfl_xor(dot, o, 32);
        cnk += __shfl_xor(cnk, o, 32);
        cnu += __shfl_xor(cnu, o, 32);
      }
      // cosine sim is invariant under the lazy per-column scales
      const float sim = dot / (knrm * sqrtf(cnk));
      const float er = __expf(sim);         // read softmax numerator
      const float ew = __expf(2.0f * sim);  // write softmax (TEMP=0.5)
      if (lane == 0) {
        ew_lds[li] = ew;
        cnk_lds[li] = cnk;
        cnu_lds[li] = cnu;
        atomicAdd(&sS[0], er);
        atomicAdd(&sS[1], ew);
        if (g > 0) atomicAdd(&sF, cnk);  // Frobenius^2 of updated mem_k
      }
      const float es = er / fmaxf(sqrtf(cnu), 1e-12f);  // normalized variant
#pragma unroll
      for (int i2 = 0; i2 < 8; ++i2) {
        upA[i2].x += er * nu[i2].x; upA[i2].y += er * nu[i2].y;
        upA[i2].z += er * nu[i2].z; upA[i2].w += er * nu[i2].w;
        upB[i2].x += es * nu[i2].x; upB[i2].y += es * nu[i2].y;
        upB[i2].z += es * nu[i2].z; upB[i2].w += es * nu[i2].w;
      }
    }
#pragma unroll
    for (int i2 = 0; i2 < 8; ++i2) {
      const int d = i2 * 128 + lane * 4;
      atomicAdd(&upA_lds[d + 0], upA[i2].x);
      atomicAdd(&upA_lds[d + 1], upA[i2].y);
      atomicAdd(&upA_lds[d + 2], upA[i2].z);
      atomicAdd(&upA_lds[d + 3], upA[i2].w);
      atomicAdd(&upB_lds[d + 0], upB[i2].x);
      atomicAdd(&upB_lds[d + 1], upB[i2].y);
      atomicAdd(&upB_lds[d + 2], upB[i2].z);
      atomicAdd(&upB_lds[d + 3], upB[i2].w);
    }
    __syncthreads();
    if (tid == 0) {
      atomicAdd(&Sr[q], sS[0]);
      atomicAdd(&Sw[q], sS[1]);
      if (g > 0) atomicAdd(&frob[q], sF);
    }
    for (int i = tid; i < DU; i += TPB) {
      atomicAdd(&upA_G[q * DU + i], upA_lds[i]);
      atomicAdd(&upB_G[q * DU + i], upB_lds[i]);
    }

    grid_sync(cnt, gen);

    // ---- epilogue: frob decision -> lazy scales, emit u_mem[g] ----
    const float frobT = gloadf(&frob[q]);
    const float SrT = gloadf(&Sr[q]);
    const bool scaled = frobT > 4.0f;  // ||mem_k||_F > 2
    if (tid < COLS_PER_BLK) {
      sK_lds[tid] = scaled ? 1.f / fmaxf(sqrtf(cnk_lds[tid]), 1e-12f) : 1.f;
      sU_lds[tid] = scaled ? 1.f / fmaxf(sqrtf(cnu_lds[tid]), 1e-12f) : 1.f;
    }
    if (tid == 0) Sw_sh = gloadf(&Sw[q]);
    if (blockIdx.x == 0) {
      const float inv = 1.0f / SrT;
      const float* src = (scaled ? upB_G : upA_G) + q * DU;
      for (int i = tid; i < DU; i += TPB)
        u_mem[(size_t)g * DU + i] = gloadf(&src[i]) * inv;
    }
    // next-iteration top __syncthreads orders all LDS writes above
  }
}

// ---------------- fp32 WMMA GEMM: C = act(A @ W^T + bias) ----------------
// A is [1024, K] given as two [1024,1024] halves (A0 | A1) when K==2048.
// W is [1024, K] row-major, so B[k][n] = W[n*K + k].
// Each wave computes a 16x64 strip via 4x V_WMMA_F32_16X16X4_F32 per k-step.
__global__ void __launch_bounds__(TPB) gemm_wmma(
    const float* __restrict__ A0, const float* __restrict__ A1,
    const float* __restrict__ W, const float* __restrict__ bias,
    float* __restrict__ C, int K, int doRelu) {
  const int lane = threadIdx.x & 31;
  const int wave = threadIdx.x >> 5;
  const int rowTile = blockIdx.x & 63;   // 64 row tiles of 16
  const int colGrp  = blockIdx.x >> 6;   // 2 column groups of 512
  const int m0 = rowTile * 16;
  const int n0 = colGrp * 512 + wave * 64;
  const int r  = lane & 15;
  const int hi = lane >> 4;              // A/B fragment k-half select
  const int aRow = m0 + r;

  v8f acc[4] = {};
  for (int k0 = 0; k0 < K; k0 += 4) {
    const int kk = k0 + 2 * hi;
    const float* abase = (kk < 1024) ? A0 : A1;
    v2f a = *(const v2f*)(abase + (size_t)aRow * 1024 + (kk & 1023));
#pragma unroll
    for (int j = 0; j < 4; ++j) {
      const int n = n0 + j * 16 + r;
      v2f b = *(const v2f*)(W + (size_t)n * K + kk);
      acc[j] = __builtin_amdgcn_wmma_f32_16x16x4_f32(
          false, a, false, b, (short)0, acc[j], false, false);
    }
  }
#pragma unroll
  for (int j = 0; j < 4; ++j) {
    const int n = n0 + j * 16 + r;
    const float bv = bias[n];
#pragma unroll
    for (int rr = 0; rr < 8; ++rr) {
      const int row = m0 + rr + 8 * hi;
      float v = acc[j][rr] + bv;
      if (doRelu) v = fmaxf(v, 0.f);
      C[(size_t)row * 1024 + n] = v;
    }
  }
}

// ---------------- host launcher ----------------
extern "C" void kernel_launch(void* const* d_in, const int* in_sizes, int n_in,
                              void* d_out, int out_size, void* d_ws, size_t ws_size,
                              hipStream_t stream) {
  const float* k   = (const float*)d_in[0];
  const float* u   = (const float*)d_in[1];
  const float* mk  = (const float*)d_in[2];  // [Dk, M]
  const float* mu  = (const float*)d_in[3];  // [Du, M]
  const float* W1  = (const float*)d_in[4];  // [Du, 2*Du]
  const float* b1  = (const float*)d_in[5];
  const float* W2  = (const float*)d_in[6];  // [Du, Du]
  const float* b2  = (const float*)d_in[7];
  float* out = (float*)d_out;

  float* ws = (float*)d_ws;
  float* memKT = ws;                                   // [M, Dk]  8M floats
  float* memUT = memKT + (size_t)MCOLS * DK;           // [M, Du]  8M floats
  float* umem  = memUT + (size_t)MCOLS * DU;           // [B, Du]  1M floats
  float* hbuf  = umem + (size_t)B_STEPS * DU;          // [B, Du]  1M floats
  float* knorm = hbuf + (size_t)B_STEPS * DU;          // [B]
  float* ctrl  = knorm + B_STEPS;                      // control block (8192 floats)
  int*   cnt   = (int*)ctrl;
  int*   gen   = cnt + 1;
  float* Sr    = ctrl + 8;    // [3]
  float* Sw    = ctrl + 12;   // [3]
  float* frob  = ctrl + 16;   // [3]
  float* upA_G = ctrl + 32;                            // [3][Du]
  float* upB_G = upA_G + 3 * DU;                       // [3][Du]

  // Deterministic state for graph replay: zero barriers + accumulators.
  hipMemsetAsync(ctrl, 0, 8192 * sizeof(float), stream);

  // Stage memories transposed into workspace (inputs stay pristine).
  {
    const size_t total = (size_t)DK * MCOLS;
    const int nb = (int)((total + TPB - 1) / TPB);
    transpose_kernel<<<nb, TPB, 0, stream>>>(mk, memKT, DK, MCOLS);
    transpose_kernel<<<nb, TPB, 0, stream>>>(mu, memUT, DU, MCOLS);
  }
  knorm_kernel<<<B_STEPS, TPB, 0, stream>>>(k, knorm);

  // Sequential memory scan: persistent grid, L2-resident matrices,
  // one fused memory pass + one grid barrier per step.
  scan_kernel<<<NBLK, TPB, 0, stream>>>(k, u, memKT, memUT, umem, knorm,
                                        cnt, gen, Sr, Sw, frob, upA_G, upB_G);

  // MLP: h = relu([u, umem] @ W1^T + b1); out = h @ W2^T + b2  (fp32 WMMA)
  gemm_wmma<<<128, TPB, 0, stream>>>(u, umem, W1, b1, hbuf, 2 * DU, 1);
  gemm_wmma<<<128, TPB, 0, stream>>>(hbuf, hbuf, W2, b2, out, DU, 0);
}